// KGEncoder_21620865368733
// MI455X (gfx1250) — compile-verified
//
#include <hip/hip_runtime.h>

#define N_NODES   200000
#define NODE_DIM  256
#define HIDDEN    128
#define N_EDGES   6400000
#define N_DRUGS   1024
#define MAX_NEIGH 64
#define HSIZE     2048
#define HMASK     (HSIZE - 1)
#define CHUNK     4096
#define NCHUNKS   ((N_EDGES + CHUNK - 1) / CHUNK)   /* 1563 */
#define WPB       8                                  /* waves per block */

typedef __attribute__((ext_vector_type(2))) float v2f;
typedef __attribute__((ext_vector_type(8))) float v8f;

__device__ __forceinline__ unsigned hash_node(int v) {
    return (((unsigned)v * 2654435761u) >> 21) & HMASK;   // top 11 bits
}

__device__ __forceinline__ int hash_lookup(const int* __restrict__ hkeys,
                                           const int* __restrict__ hvals, int node) {
    unsigned h = hash_node(node);
    while (true) {
        int k = hkeys[h];
        if (k == node) return hvals[h];
        if (k == -1) return -1;
        h = (h + 1) & HMASK;
    }
}

// ---- hash build ------------------------------------------------------------
__global__ void k_init_hash(int* __restrict__ hkeys) {
    int t = blockIdx.x * blockDim.x + threadIdx.x;
    if (t < HSIZE) hkeys[t] = -1;
}

__global__ void k_insert(const int* __restrict__ drug_nodes,
                         int* __restrict__ hkeys, int* __restrict__ hvals) {
    int d = threadIdx.x;                      // 1024 threads, one block
    int node = drug_nodes[d];
    unsigned h = hash_node(node);
    while (true) {
        int old = atomicCAS(&hkeys[h], -1, node);
        if (old == -1) { hvals[h] = d; break; }   // I own this slot
        if (old == node) break;                   // duplicate node id
        h = (h + 1) & HMASK;
    }
}

__global__ void k_slot(const int* __restrict__ drug_nodes,
                       const int* __restrict__ hkeys, const int* __restrict__ hvals,
                       int* __restrict__ slot_of_drug) {
    int d = threadIdx.x;                      // 1024 threads, one block
    slot_of_drug[d] = hash_lookup(hkeys, hvals, drug_nodes[d]);
}

// ---- pass 1: per-chunk per-slot match counts (one wave per chunk) ----------
__global__ void __launch_bounds__(256) k_count(const int* __restrict__ src,
                       const int* __restrict__ hkeys, const int* __restrict__ hvals,
                       int* __restrict__ counts) {
    __shared__ int cnt[WPB * N_DRUGS];
    int wave = threadIdx.x >> 5;
    int lane = threadIdx.x & 31;
    int chunk = blockIdx.x * WPB + wave;
    if (chunk >= NCHUNKS) return;
    int* c = cnt + wave * N_DRUGS;
    for (int j = lane; j < N_DRUGS; j += 32) c[j] = 0;
    int base = chunk * CHUNK;
    for (int i = 0; i < CHUNK; i += 32) {
        int e = base + i + lane;
        if (e < N_EDGES) {
            int slot = hash_lookup(hkeys, hvals, src[e]);
            if (slot >= 0) atomicAdd(&c[slot], 1);
        }
    }
    for (int j = lane; j < N_DRUGS; j += 32)
        counts[(size_t)chunk * N_DRUGS + j] = c[j];
}

// ---- pass 2: exclusive prefix over chunks, per slot ------------------------
__global__ void k_prefix(int* __restrict__ counts, int* __restrict__ totals) {
    int j = blockIdx.x * blockDim.x + threadIdx.x;
    if (j >= N_DRUGS) return;
    int run = 0;
    for (int c = 0; c < NCHUNKS; ++c) {
        size_t idx = (size_t)c * N_DRUGS + j;
        int t = counts[idx];
        counts[idx] = run;        // exclusive offset
        run += t;
    }
    totals[j] = run;
}

// ---- pass 3: rank assignment in exact edge order (one wave per chunk) ------
__global__ void __launch_bounds__(256) k_fill(const int* __restrict__ src,
                      const int* __restrict__ dst,
                      const int* __restrict__ hkeys, const int* __restrict__ hvals,
                      const int* __restrict__ offs, int* __restrict__ neigh) {
    __shared__ int cnt[WPB * N_DRUGS];
    int wave = threadIdx.x >> 5;
    int lane = threadIdx.x & 31;
    int chunk = blockIdx.x * WPB + wave;
    if (chunk >= NCHUNKS) return;
    int* c = cnt + wave * N_DRUGS;
    for (int j = lane; j < N_DRUGS; j += 32) c[j] = 0;
    int base = chunk * CHUNK;
    for (int i = 0; i < CHUNK; i += 32) {
        int e = base + i + lane;
        int slot = -1;
        if (e < N_EDGES) slot = hash_lookup(hkeys, hvals, src[e]);
        unsigned m = (unsigned)__ballot(slot >= 0);
        while (m) {                         // serialize matches, lowest lane first
            int b = __builtin_ctz(m);
            if (lane == b) {
                int loc = c[slot];
                c[slot] = loc + 1;
                int r = offs[(size_t)chunk * N_DRUGS + slot] + loc;
                if (r < MAX_NEIGH) neigh[slot * MAX_NEIGH + r] = dst[e];
            }
            m &= m - 1;
        }
    }
}

// ---- gather indices + mask -------------------------------------------------
__global__ void k_gather(const int* __restrict__ slot_of_drug,
                         const int* __restrict__ totals,
                         const int* __restrict__ neigh,
                         const int* __restrict__ drug_nodes,
                         int* __restrict__ gidx, float* __restrict__ maskout) {
    int t = blockIdx.x * blockDim.x + threadIdx.x;   // 65536 threads
    int d = t >> 6, k = t & 63;
    int slot = slot_of_drug[d];
    int total = totals[slot];
    int cn = total < MAX_NEIGH ? total : MAX_NEIGH;
    int g; float mv;
    if (k < cn)                      { g = neigh[slot * MAX_NEIGH + k]; mv = 1.0f; }
    else if (total == 0 && k == 0)   { g = drug_nodes[d];               mv = 1.0f; }
    else                             { g = 0;                           mv = 0.0f; }
    gidx[t] = g;
    maskout[t] = mv;
}

// ---- WMMA fp32 GEMM over gathered rows: out = (X[g] @ W + b) * mask --------
// One wave computes one 16(M)x16(N) tile; K-loop over 256 in steps of 4.
__global__ void __launch_bounds__(256) k_gemm(
        const float* __restrict__ X, const float* __restrict__ W,
        const float* __restrict__ bias, const int* __restrict__ gidx,
        const float* __restrict__ maskv, float* __restrict__ out) {
    int wave = threadIdx.x >> 5;
    int lane = threadIdx.x & 31;
    int wid  = blockIdx.x * WPB + wave;     // 32768 waves total (exact)
    int rt = wid >> 3;                      // 4096 row tiles of 16
    int ct = wid & 7;                       // 8 col tiles of 16
    int rsel = lane & 15;
    int grp  = lane >> 4;

    // A (16x4 fp32): lanes 0-15 hold K={k0,k0+1}, lanes 16-31 hold K={k0+2,k0+3}
    int r = rt * 16 + rsel;
    int g = gidx[r];
    const float* ap = X + (size_t)g * NODE_DIM + (grp << 1);
    // B (4x16 fp32): lanes 0-15: rows {k0,k0+1}; lanes 16-31: rows {k0+2,k0+3}
    int n = ct * 16 + rsel;
    const float* bp = W + (size_t)(grp << 1) * HIDDEN + n;

    v8f acc = {};
    #pragma unroll 4
    for (int k0 = 0; k0 < NODE_DIM; k0 += 4) {
        v2f a = *(const v2f*)(ap + k0);
        v2f bv;
        bv.x = bp[(size_t)k0 * HIDDEN];
        bv.y = bp[(size_t)(k0 + 1) * HIDDEN];
        acc = __builtin_amdgcn_wmma_f32_16x16x4_f32(
                  false, a, false, bv, (short)0, acc, false, false);
    }
    float bb = bias[n];
    #pragma unroll
    for (int i = 0; i < 8; ++i) {
        int rr = rt * 16 + (grp << 3) + i;      // C/D layout: VGPR i -> M = i + 8*grp
        out[(size_t)rr * HIDDEN + n] = (acc[i] + bb) * maskv[rr];
    }
}

extern "C" void kernel_launch(void* const* d_in, const int* in_sizes, int n_in,
                              void* d_out, int out_size, void* d_ws, size_t ws_size,
                              hipStream_t stream) {
    const float* node_feats = (const float*)d_in[0];
    const float* W          = (const float*)d_in[1];
    const float* bias       = (const float*)d_in[2];
    const int*   edge_index = (const int*)d_in[3];
    /* d_in[4] = edge_type: unused by the reference output */
    const int*   drug_nodes = (const int*)d_in[5];
    const int* src = edge_index;
    const int* dst = edge_index + N_EDGES;

    char* ws = (char*)d_ws;
    int* hkeys        = (int*)ws;  ws += HSIZE * 4;
    int* hvals        = (int*)ws;  ws += HSIZE * 4;
    int* slot_of_drug = (int*)ws;  ws += N_DRUGS * 4;
    int* totals       = (int*)ws;  ws += N_DRUGS * 4;
    int* neigh        = (int*)ws;  ws += N_DRUGS * MAX_NEIGH * 4;
    int* gidx         = (int*)ws;  ws += N_DRUGS * MAX_NEIGH * 4;
    int* counts       = (int*)ws;  /* NCHUNKS * N_DRUGS * 4 bytes ~ 6.4 MB */

    float* out     = (float*)d_out;
    float* maskout = out + (size_t)N_DRUGS * MAX_NEIGH * HIDDEN;

    int cblocks = (NCHUNKS + WPB - 1) / WPB;

    k_init_hash<<<(HSIZE + 255) / 256, 256, 0, stream>>>(hkeys);
    k_insert<<<1, N_DRUGS, 0, stream>>>(drug_nodes, hkeys, hvals);
    k_slot<<<1, N_DRUGS, 0, stream>>>(drug_nodes, hkeys, hvals, slot_of_drug);
    k_count<<<cblocks, 256, 0, stream>>>(src, hkeys, hvals, counts);
    k_prefix<<<(N_DRUGS + 255) / 256, 256, 0, stream>>>(counts, totals);
    k_fill<<<cblocks, 256, 0, stream>>>(src, dst, hkeys, hvals, counts, neigh);
    k_gather<<<(N_DRUGS * MAX_NEIGH) / 256, 256, 0, stream>>>(
        slot_of_drug, totals, neigh, drug_nodes, gidx, maskout);
    k_gemm<<<(N_DRUGS * MAX_NEIGH / 16) * 8 / WPB, 256, 0, stream>>>(
        node_feats, W, bias, gidx, maskout, out);
}